// LengthRegulator_17781164605703
// MI455X (gfx1250) — compile-verified
//
#include <hip/hip_runtime.h>
#include <hip/hip_bf16.h>

// LengthRegulator: out[b, j, :] = x[b, searchsorted_right(cum[b], j), :] for j < tgt_len[b], else 0
//   x:        [B, T, D] f32
//   duration: [B, T] i32   (masked by pos < src_lens[b])
//   src_lens: [B] i32
//   out:      [B, L, D] f32  ++  tgt_len [B] i32 (concatenated flat in d_out)
//
// Memory-bound: ~128 MiB NT stores + ~16 MiB gathered loads (L2-resident).
// CDNA5 paths used: global_load_async_to_lds_b128 + s_wait_asynccnt for the
// cumsum tile, LDS scan/search, b128 vector moves, TH=NT output stores.

typedef float __attribute__((ext_vector_type(4))) f4;
typedef int   __attribute__((ext_vector_type(4))) v4i;

#define LR_FPB 64   // frames per block in the expand kernel

#if defined(__gfx1250__)
#ifdef __has_builtin
#if __has_builtin(__builtin_amdgcn_global_load_async_to_lds_b128)
#define LR_ASYNC_LDS 1
#endif
#endif
#endif

// ---------------------------------------------------------------------------
// Kernel A: per-batch masked inclusive scan of durations (T threads per block)
// ---------------------------------------------------------------------------
__global__ void lr_scan_kernel(const int* __restrict__ duration,
                               const int* __restrict__ src_lens,
                               int* __restrict__ cum,      // [B*T] workspace
                               int* __restrict__ tgt_len,  // [B] (tail of d_out)
                               int T) {
    extern __shared__ int s[];  // T ints
    const int b = blockIdx.x;
    const int t = threadIdx.x;
    const int sl = src_lens[b];
    s[t] = (t < sl) ? duration[(long long)b * T + t] : 0;
    __syncthreads();
    // Hillis-Steele inclusive scan
    for (int off = 1; off < T; off <<= 1) {
        int add = (t >= off) ? s[t - off] : 0;
        __syncthreads();
        s[t] += add;
        __syncthreads();
    }
    cum[(long long)b * T + t] = s[t];
    if (t == T - 1) tgt_len[b] = s[t];
}

// ---------------------------------------------------------------------------
// Kernel B: expand. 256 threads = 8 wave32s; LR_FPB frames per block.
// Each wave handles LR_FPB/8 frames: uniform binary search on the LDS cumsum
// row, then lane-parallel b128 row copy (or zero-fill).
// ---------------------------------------------------------------------------
__global__ void __launch_bounds__(256)
lr_expand_kernel(const float* __restrict__ x,
                 const int* __restrict__ cum,
                 float* __restrict__ out,
                 int T, int D, int L, int chunksPerB) {
    extern __shared__ int s_cum[];  // T ints
    const int b      = blockIdx.x / chunksPerB;
    const int frame0 = (blockIdx.x % chunksPerB) * LR_FPB;
    const int tid    = threadIdx.x;
    const int* gcum  = cum + (long long)b * T;

#ifdef LR_ASYNC_LDS
    {
        const int nvec = T >> 2;  // 16-byte chunks
        if (tid < nvec) {
            __builtin_amdgcn_global_load_async_to_lds_b128(
                (__attribute__((address_space(1))) v4i*)(gcum + tid * 4),
                (__attribute__((address_space(3))) v4i*)(&s_cum[tid * 4]),
                /*offset=*/0, /*cpol=*/0);
        }
#if __has_builtin(__builtin_amdgcn_s_wait_asynccnt)
        __builtin_amdgcn_s_wait_asynccnt(0);
#else
        asm volatile("s_wait_asynccnt 0" ::: "memory");
#endif
    }
#else
    for (int i = tid; i < T; i += (int)blockDim.x) s_cum[i] = gcum[i];
#endif
    __syncthreads();

    const int tgt  = s_cum[T - 1];
    const int wave = tid >> 5;
    const int lane = tid & 31;
    const int framesPerWave = LR_FPB >> 3;  // 8 waves per block

    for (int k = 0; k < framesPerWave; ++k) {
        const int j = frame0 + wave * framesPerWave + k;
        if (j >= L) break;
        f4* dst = (f4*)(out + ((long long)b * L + j) * D);
        if (j < tgt) {
            // searchsorted(cum, j, side='right'): first i with cum[i] > j
            int lo = 0, hi = T;
            while (lo < hi) {
                const int mid = (lo + hi) >> 1;
                if (s_cum[mid] <= j) lo = mid + 1; else hi = mid;
            }
            const int idx = (lo < T) ? lo : (T - 1);
            const f4* src = (const f4*)(x + ((long long)b * T + idx) * D);
            for (int c = lane; c * 4 < D; c += 32) {
                f4 v = src[c];                          // L2-resident gather
                __builtin_nontemporal_store(v, &dst[c]); // streaming NT store
            }
        } else {
            const f4 z = {0.f, 0.f, 0.f, 0.f};
            for (int c = lane; c * 4 < D; c += 32) {
                __builtin_nontemporal_store(z, &dst[c]);
            }
        }
    }
}

// ---------------------------------------------------------------------------
extern "C" void kernel_launch(void* const* d_in, const int* in_sizes, int n_in,
                              void* d_out, int out_size, void* d_ws, size_t ws_size,
                              hipStream_t stream) {
    const float* x        = (const float*)d_in[0];
    const int*   duration = (const int*)d_in[1];
    const int*   src_lens = (const int*)d_in[2];
    // d_in[3] = max_len scalar on device; L is derived from out_size instead.

    const int B  = in_sizes[2];
    const int BT = in_sizes[1];
    const int T  = BT / B;                 // 512
    const int D  = in_sizes[0] / BT;       // 256
    const long long L =
        (((long long)out_size) - B) / ((long long)B * D);  // 4096 (max_len)

    float* out = (float*)d_out;
    int*   tgt = (int*)(out + (long long)B * L * D);  // tgt_len tail (int bits)
    int*   cum = (int*)d_ws;                          // B*T ints of scratch

    lr_scan_kernel<<<B, T, T * (int)sizeof(int), stream>>>(
        duration, src_lens, cum, tgt, T);

    const int chunksPerB = (int)((L + LR_FPB - 1) / LR_FPB);
    lr_expand_kernel<<<B * chunksPerB, 256, T * (int)sizeof(int), stream>>>(
        x, cum, out, T, D, (int)L, chunksPerB);
}